// SMRNBlock_4011499454831
// MI455X (gfx1250) — compile-verified
//
#include <hip/hip_runtime.h>
#include <math.h>

typedef __attribute__((ext_vector_type(2))) float v2f;
typedef __attribute__((ext_vector_type(8))) float v8f;

#define DM 256
#define DSTATE 16
#define SEQ 1024
#define NB 2
#define NTOK (NB * SEQ)
#define FDIM 512
#define WIN 8
#define GK 516   // gate GEMM K (513 zero-padded to multiple of 4)

enum { EPI_NONE = 0, EPI_BIAS = 1, EPI_BIAS_SILU = 2, EPI_BIAS_SIGMOID = 3,
       EPI_BIAS_GELU = 4, EPI_BIAS_SOFTPLUS = 5 };

__device__ __forceinline__ float epi_apply(float v, int epi) {
  switch (epi) {
    case EPI_BIAS_SILU:     return v / (1.f + expf(-v));                  // x*sigmoid(x)
    case EPI_BIAS_SIGMOID:  return 1.f / (1.f + expf(-v));
    case EPI_BIAS_GELU:     return 0.5f * v * (1.f + erff(v * 0.7071067811865476f));
    case EPI_BIAS_SOFTPLUS: return (v > 20.f) ? v : log1pf(expf(v));
    default:                return v;
  }
}

// ---------------------------------------------------------------------------
// fp32 WMMA GEMM, C = A(MxK) * B(KxN). One wave per (16*MT)x(16*NT) C tile:
// MT*NT independent accumulators, 1 global load per WMMA. K must be % 4 == 0.
// grid = (N/(16*NT), M/(16*MT), batch); blockIdx.z applies strides sA/sB/sC.
// A frag (16x4 f32): lanes 0-15 rows M=0..15 K={k,k+1}; lanes 16-31 K={k+2,k+3}
// B frag (4x16 f32): lanes 0-15 cols N=0..15 K={k,k+1}; lanes 16-31 K={k+2,k+3}
// C/D (16x16 f32, 8 VGPRs): vgpr r -> row r (lanes 0-15) / row r+8 (lanes 16-31)
// ---------------------------------------------------------------------------
template <int MT, int NT>
__global__ __launch_bounds__(32)
void gemm_nn_wmma(const float* __restrict__ A, const float* __restrict__ Bm,
                  const float* __restrict__ bias, float* __restrict__ C,
                  int K, int lda, int ldb, int ldc,
                  long sA, long sB, long sC, int epi) {
  const long bz = blockIdx.z;
  A += bz * sA;  Bm += bz * sB;  C += bz * sC;
  const int lane = threadIdx.x & 31;
  const int half = lane >> 4;
  const int l    = lane & 15;
  const int m0   = blockIdx.y * (16 * MT);
  const int n0   = blockIdx.x * (16 * NT);
  v8f acc[MT][NT] = {};
  for (int k = 0; k < K; k += 4) {
    const int ka = k + 2 * half;
    v2f a[MT], b[NT];
#pragma unroll
    for (int i = 0; i < MT; ++i)
      a[i] = *(const v2f*)(A + (long)(m0 + i * 16 + l) * lda + ka);
#pragma unroll
    for (int j = 0; j < NT; ++j) {
      const float* bp = Bm + (long)ka * ldb + n0 + j * 16 + l;
      b[j].x = bp[0];
      b[j].y = bp[ldb];
    }
#pragma unroll
    for (int i = 0; i < MT; ++i)
#pragma unroll
      for (int j = 0; j < NT; ++j)
        acc[i][j] = __builtin_amdgcn_wmma_f32_16x16x4_f32(
            false, a[i], false, b[j], (short)0, acc[i][j], false, false);
  }
#pragma unroll
  for (int i = 0; i < MT; ++i) {
    const int r0 = m0 + i * 16 + half * 8;
#pragma unroll
    for (int j = 0; j < NT; ++j) {
      const int col = n0 + j * 16 + l;
      const float bv = bias ? bias[col] : 0.f;
#pragma unroll
      for (int r = 0; r < 8; ++r)
        C[(long)(r0 + r) * ldc + col] = epi_apply(acc[i][j][r] + bv, epi);
    }
  }
}

// C = A(MxK) * B(NxK)^T with causal mask (col > row -> 0). B loads contiguous.
template <int MT, int NT>
__global__ __launch_bounds__(32)
void gemm_nt_causal(const float* __restrict__ A, const float* __restrict__ Bm,
                    float* __restrict__ C, int K, int lda, int ldb, int ldc,
                    long sA, long sB, long sC) {
  const long bz = blockIdx.z;
  A += bz * sA;  Bm += bz * sB;  C += bz * sC;
  const int lane = threadIdx.x & 31;
  const int half = lane >> 4;
  const int l    = lane & 15;
  const int m0   = blockIdx.y * (16 * MT);
  const int n0   = blockIdx.x * (16 * NT);
  v8f acc[MT][NT] = {};
  for (int k = 0; k < K; k += 4) {
    const int ka = k + 2 * half;
    v2f a[MT], b[NT];
#pragma unroll
    for (int i = 0; i < MT; ++i)
      a[i] = *(const v2f*)(A + (long)(m0 + i * 16 + l) * lda + ka);
#pragma unroll
    for (int j = 0; j < NT; ++j)
      b[j] = *(const v2f*)(Bm + (long)(n0 + j * 16 + l) * ldb + ka);
#pragma unroll
    for (int i = 0; i < MT; ++i)
#pragma unroll
      for (int j = 0; j < NT; ++j)
        acc[i][j] = __builtin_amdgcn_wmma_f32_16x16x4_f32(
            false, a[i], false, b[j], (short)0, acc[i][j], false, false);
  }
#pragma unroll
  for (int i = 0; i < MT; ++i) {
    const int r0 = m0 + i * 16 + half * 8;
#pragma unroll
    for (int j = 0; j < NT; ++j) {
      const int col = n0 + j * 16 + l;
#pragma unroll
      for (int r = 0; r < 8; ++r) {
        const int grow = r0 + r;
        C[(long)grow * ldc + col] = (col <= grow) ? acc[i][j][r] : 0.f;
      }
    }
  }
}

// ---------------------------------------------------------------------------
// SSM selective scan: grid = NB blocks x DM threads; h[16] in registers.
// ---------------------------------------------------------------------------
__global__ __launch_bounds__(DM)
void ssm_scan(const float* __restrict__ x, const float* __restrict__ A_log,
              const float* __restrict__ delta, const float* __restrict__ xp,
              const float* __restrict__ ct, float* __restrict__ y_ssm) {
  const int b = blockIdx.x;
  const int d = threadIdx.x;
  float Aneg[DSTATE], h[DSTATE];
  for (int n = 0; n < DSTATE; ++n) { Aneg[n] = -expf(A_log[d * DSTATE + n]); h[n] = 0.f; }
  __shared__ float sB[DSTATE], sC[DSTATE];
  for (int s = 0; s < SEQ; ++s) {
    const long t = (long)b * SEQ + s;
    if (d < DSTATE)           sB[d]          = xp[t * 32 + DSTATE + d];
    else if (d < 2 * DSTATE)  sC[d - DSTATE] = ct[t * DSTATE + (d - DSTATE)];
    __syncthreads();
    const float dlt = delta[t * DM + d];
    const float dx  = dlt * x[t * DM + d];
    float y = 0.f;
    for (int n = 0; n < DSTATE; ++n) {
      const float ab = expf(dlt * Aneg[n]);
      h[n] = ab * h[n] + dx * sB[n];
      y += sC[n] * h[n];
    }
    y_ssm[t * DM + d] = y;
    __syncthreads();
  }
}

// ---------------------------------------------------------------------------
// Small elementwise / reduction kernels
// ---------------------------------------------------------------------------
__global__ void rff_kernel(const float* pq, const float* pk, float* qp, float* kp, int n) {
  int i = blockIdx.x * blockDim.x + threadIdx.x;
  if (i >= n) return;
  const int t = i / DM, c = i % DM;
  const float sc = 0.08838834764831845f;  // sqrt(2/256)
  qp[(long)t * FDIM + c]      = sc * cosf(pq[i]);
  qp[(long)t * FDIM + DM + c] = sc * sinf(pq[i]);
  kp[(long)t * FDIM + c]      = sc * cosf(pk[i]);
  kp[(long)t * FDIM + DM + c] = sc * sinf(pk[i]);
}

__global__ void rowsum_kernel(const float* sc, float* den, int n) {
  int t = blockIdx.x * blockDim.x + threadIdx.x;
  if (t >= n) return;
  const int b = t >> 10, s = t & (SEQ - 1);
  const float* row = sc + (long)b * SEQ * SEQ + (long)s * SEQ;
  float acc = 0.f;
  for (int j = 0; j < SEQ; ++j) acc += row[j];
  den[t] = acc;
}

__global__ void div_kernel(const float* num, const float* den, float* out, int n) {
  int i = blockIdx.x * blockDim.x + threadIdx.x;
  if (i >= n) return;
  out[i] = num[i] / (den[i / DM] + 1e-6f);
}

__global__ void norm_kernel(const float* x, float* m, int n) {
  int t = blockIdx.x * blockDim.x + threadIdx.x;
  if (t >= n) return;
  float acc = 0.f;
  for (int c = 0; c < DM; ++c) { float v = x[(long)t * DM + c]; acc += v * v; }
  m[t] = sqrtf(acc);
}

__global__ void entropy_kernel(const float* m, float* H, int n) {
  int t = blockIdx.x * blockDim.x + threadIdx.x;
  if (t >= n) return;
  const int b = t >> 10, s = t & (SEQ - 1);
  float w[WIN], mx = -1e30f;
  for (int i = 0; i < WIN; ++i) {
    const int idx = s - (WIN - 1) + i;
    w[i] = (idx >= 0) ? m[b * SEQ + idx] : -INFINITY;
    if (w[i] > mx) mx = w[i];
  }
  float sum = 0.f;
  for (int i = 0; i < WIN; ++i) { w[i] = expf(w[i] - mx); sum += w[i]; }
  float h = 0.f;
  const float inv_ln2 = 1.4426950408889634f;
  for (int i = 0; i < WIN; ++i) {
    const float p = w[i] / sum;
    h -= p * logf(p + 1e-9f) * inv_ln2;
  }
  H[t] = h;
}

// gate_in padded to GK columns: [y_ssm(256) | y_attn(256) | H(1) | zeros(3)]
__global__ void concat_kernel(const float* y_ssm, const float* y_attn, const float* H,
                              float* gate_in, int n) {
  int i = blockIdx.x * blockDim.x + threadIdx.x;
  if (i >= n) return;
  const int t = i / GK, c = i % GK;
  float v;
  if (c < DM)            v = y_ssm[(long)t * DM + c];
  else if (c < 2 * DM)   v = y_attn[(long)t * DM + (c - DM)];
  else if (c == 2 * DM)  v = H[t];
  else                   v = 0.f;
  gate_in[i] = v;
}

// zero-pad Wg1 (513 x 256) to (GK x 256)
__global__ void padw_kernel(const float* Wg1, float* Wg1p, int n) {
  int i = blockIdx.x * blockDim.x + threadIdx.x;
  if (i >= n) return;
  const int r = i / DM;
  Wg1p[i] = (r < 2 * DM + 1) ? Wg1[i] : 0.f;
}

__global__ void gated_kernel(const float* g, const float* y_ssm, const float* y_attn,
                             float* yg, int n) {
  int i = blockIdx.x * blockDim.x + threadIdx.x;
  if (i >= n) return;
  const float gv = g[i];
  yg[i] = gv * y_ssm[i] + (1.f - gv) * y_attn[i];
}

// out = LN(a + b_opt), one block (DM threads) per token
__global__ __launch_bounds__(DM)
void ln_kernel(const float* __restrict__ a, const float* __restrict__ bopt,
               const float* __restrict__ g, const float* __restrict__ beta,
               float* __restrict__ out) {
  const long t = blockIdx.x;
  const int i = threadIdx.x;
  float v = a[t * DM + i] + (bopt ? bopt[t * DM + i] : 0.f);
  __shared__ float red[DM];
  red[i] = v; __syncthreads();
  for (int st = DM / 2; st > 0; st >>= 1) { if (i < st) red[i] += red[i + st]; __syncthreads(); }
  const float mu = red[0] * (1.f / DM);
  __syncthreads();
  const float dv = v - mu;
  red[i] = dv * dv; __syncthreads();
  for (int st = DM / 2; st > 0; st >>= 1) { if (i < st) red[i] += red[i + st]; __syncthreads(); }
  const float var = red[0] * (1.f / DM);
  out[t * DM + i] = dv * rsqrtf(var + 1e-5f) * g[i] + beta[i];
}

// ---------------------------------------------------------------------------
extern "C" void kernel_launch(void* const* d_in, const int* in_sizes, int n_in,
                              void* d_out, int out_size, void* d_ws, size_t ws_size,
                              hipStream_t stream) {
  const float* x       = (const float*)d_in[0];
  const float* A_log   = (const float*)d_in[1];
  const float* W_xproj = (const float*)d_in[2];
  const float* W_dt    = (const float*)d_in[3];
  const float* b_dt    = (const float*)d_in[4];
  const float* W_C     = (const float*)d_in[5];
  const float* Wq      = (const float*)d_in[6];
  const float* Wk      = (const float*)d_in[7];
  const float* Wv      = (const float*)d_in[8];
  const float* Wo      = (const float*)d_in[9];
  const float* omega   = (const float*)d_in[10];
  const float* Wg1     = (const float*)d_in[11];
  const float* bg1     = (const float*)d_in[12];
  const float* Wg2     = (const float*)d_in[13];
  const float* bg2     = (const float*)d_in[14];
  const float* ln1_g   = (const float*)d_in[15];
  const float* ln1_b   = (const float*)d_in[16];
  const float* ln2_g   = (const float*)d_in[17];
  const float* ln2_b   = (const float*)d_in[18];
  const float* Wf1     = (const float*)d_in[19];
  const float* bf1     = (const float*)d_in[20];
  const float* Wf2     = (const float*)d_in[21];
  const float* bf2     = (const float*)d_in[22];
  float* out = (float*)d_out;

  float* p = (float*)d_ws;
  auto nxt = [&](size_t nf) { float* r = p; p += nf; return r; };
  const size_t MD = (size_t)NTOK * DM;
  float* xq      = nxt(MD);
  float* xk      = nxt(MD);
  float* xv      = nxt(MD);
  float* pq      = nxt(MD);
  float* pk      = nxt(MD);
  float* qp      = nxt((size_t)NTOK * FDIM);
  float* kp      = nxt((size_t)NTOK * FDIM);
  float* scores  = nxt((size_t)NB * SEQ * SEQ);
  float* ynum    = nxt(MD);
  float* den     = nxt(NTOK);
  float* yin     = nxt(MD);
  float* y_attn  = nxt(MD);
  float* xp      = nxt((size_t)NTOK * 32);
  float* delta   = nxt(MD);
  float* ct      = nxt((size_t)NTOK * DSTATE);
  float* y_ssm   = nxt(MD);
  float* mnorm   = nxt(NTOK);
  float* Hent    = nxt(NTOK);
  float* gate_in = nxt((size_t)NTOK * GK);
  float* Wg1p    = nxt((size_t)GK * DM);
  float* g1      = nxt(MD);
  float* gbuf    = nxt(MD);
  float* ygated  = nxt(MD);
  float* hbuf    = nxt(MD);
  float* ffn1    = nxt((size_t)NTOK * 4 * DM);
  float* ffn2    = nxt(MD);

  // 64x32 tile per wave (MT=4, NT=2)
  auto gemm = [&](const float* A, const float* B, const float* bias, float* C,
                  int M, int N, int K, int lda, int ldb, int ldc, int epi,
                  int nb = 1, long sA = 0, long sB = 0, long sC = 0) {
    gemm_nn_wmma<4, 2><<<dim3(N / 32, M / 64, nb), dim3(32), 0, stream>>>(
        A, B, bias, C, K, lda, ldb, ldc, sA, sB, sC, epi);
  };

  // --- token-parallel projections (fp32 WMMA) ---
  gemm(x,  Wq,      nullptr, xq, NTOK, DM, DM, DM, DM, DM, EPI_NONE);
  gemm(x,  Wk,      nullptr, xk, NTOK, DM, DM, DM, DM, DM, EPI_NONE);
  gemm(x,  Wv,      nullptr, xv, NTOK, DM, DM, DM, DM, DM, EPI_NONE);
  gemm(xq, omega,   nullptr, pq, NTOK, DM, DM, DM, DM, DM, EPI_NONE);
  gemm(xk, omega,   nullptr, pk, NTOK, DM, DM, DM, DM, DM, EPI_NONE);
  gemm(x,  W_xproj, nullptr, xp, NTOK, 32, DM, DM, 32, 32, EPI_NONE);
  gemm(xp, W_dt,    b_dt, delta, NTOK, DM, DSTATE, 32, DM, DM, EPI_BIAS_SOFTPLUS);
  gemm_nn_wmma<4, 1><<<dim3(1, NTOK / 64, 1), dim3(32), 0, stream>>>(   // N=16
      x, W_C, nullptr, ct, DM, DM, DSTATE, DSTATE, 0, 0, 0, EPI_NONE);

  // --- SSM selective scan (serial over S, parallel over B*d channels) ---
  ssm_scan<<<dim3(NB), dim3(DM), 0, stream>>>(x, A_log, delta, xp, ct, y_ssm);

  // --- linear attention == causal kernel attention (no serial scan) ---
  const int EW = 256;
  rff_kernel<<<dim3((NTOK * DM + EW - 1) / EW), dim3(EW), 0, stream>>>(pq, pk, qp, kp, NTOK * DM);
  gemm_nt_causal<4, 2><<<dim3(SEQ / 32, SEQ / 64, NB), dim3(32), 0, stream>>>(
      qp, kp, scores, FDIM, FDIM, FDIM, SEQ,
      (long)SEQ * FDIM, (long)SEQ * FDIM, (long)SEQ * SEQ);
  rowsum_kernel<<<dim3((NTOK + EW - 1) / EW), dim3(EW), 0, stream>>>(scores, den, NTOK);
  gemm(scores, xv, nullptr, ynum, SEQ, DM, SEQ, SEQ, DM, DM, EPI_NONE,
       NB, (long)SEQ * SEQ, (long)SEQ * DM, (long)SEQ * DM);
  div_kernel<<<dim3((NTOK * DM + EW - 1) / EW), dim3(EW), 0, stream>>>(ynum, den, yin, NTOK * DM);
  gemm(yin, Wo, nullptr, y_attn, NTOK, DM, DM, DM, DM, DM, EPI_NONE);

  // --- windowed entropy of token norms ---
  norm_kernel<<<dim3((NTOK + EW - 1) / EW), dim3(EW), 0, stream>>>(x, mnorm, NTOK);
  entropy_kernel<<<dim3((NTOK + EW - 1) / EW), dim3(EW), 0, stream>>>(mnorm, Hent, NTOK);

  // --- gating MLP (K padded 513 -> 516 so the WMMA K-loop has no tail) ---
  concat_kernel<<<dim3((NTOK * GK + EW - 1) / EW), dim3(EW), 0, stream>>>(
      y_ssm, y_attn, Hent, gate_in, NTOK * GK);
  padw_kernel<<<dim3((GK * DM + EW - 1) / EW), dim3(EW), 0, stream>>>(Wg1, Wg1p, GK * DM);
  gemm(gate_in, Wg1p, bg1, g1,   NTOK, DM, GK, GK, DM, DM, EPI_BIAS_SILU);
  gemm(g1,      Wg2,  bg2, gbuf, NTOK, DM, DM, DM, DM, DM, EPI_BIAS_SIGMOID);
  gated_kernel<<<dim3((NTOK * DM + EW - 1) / EW), dim3(EW), 0, stream>>>(
      gbuf, y_ssm, y_attn, ygated, NTOK * DM);

  // --- residual + LN1, FFN (GELU), residual + LN2 ---
  ln_kernel<<<dim3(NTOK), dim3(DM), 0, stream>>>(x, ygated, ln1_g, ln1_b, hbuf);
  gemm(hbuf, Wf1, bf1, ffn1, NTOK, 4 * DM, DM, DM, 4 * DM, 4 * DM, EPI_BIAS_GELU);
  gemm(ffn1, Wf2, bf2, ffn2, NTOK, DM, 4 * DM, 4 * DM, DM, DM, EPI_BIAS);
  ln_kernel<<<dim3(NTOK), dim3(DM), 0, stream>>>(hbuf, ffn2, ln2_g, ln2_b, out);
}